// CholeskyConstraintLayer_6141803233689
// MI455X (gfx1250) — compile-verified
//
#include <hip/hip_runtime.h>

typedef __attribute__((ext_vector_type(2))) float v2f;
typedef __attribute__((ext_vector_type(8))) float v8f;

#define WAVES_PER_BLOCK 8
#define ELEMS_PER_WAVE  4
#define ELEMS_PER_BLOCK (WAVES_PER_BLOCK * ELEMS_PER_WAVE) // 32

// Wave-local LDS fence: all LDS traffic here is private to the wave, and
// same-wave DS ops complete in order (CDNA5 ISA 7.3), so a block barrier is
// overkill -- just drain DScnt and stop compiler reordering.
__device__ __forceinline__ void wave_lds_fence() {
    __builtin_amdgcn_wave_barrier();
    asm volatile("s_wait_dscnt 0x0" ::: "memory");
    __builtin_amdgcn_wave_barrier();
}

// Each wave computes rho = L L^H for 4 batch elements with V_WMMA_F32_16X16X4_F32.
// A (16x4) = four 4x4 L stacked by rows; B (4x16) = four L^T side by side;
// the diagonal 4x4 blocks of D are the per-element products.
__global__ __launch_bounds__(256)
void cholesky_rho_wmma_kernel(const float* __restrict__ x,
                              float* __restrict__ out, int B) {
    // +16 pad: phase-2 reads idx+1 unconditionally (value discarded when c>=t)
    __shared__ float lds_in [WAVES_PER_BLOCK * 64 + 16];
    // full 16x16 D tile of (re,im) per wave: 512 floats / wave
    __shared__ float lds_out[WAVES_PER_BLOCK * 512];

    const int tid    = threadIdx.x;
    const int wave   = tid >> 5;
    const int lane   = tid & 31;
    const int base_e = (blockIdx.x * WAVES_PER_BLOCK + wave) * ELEMS_PER_WAVE;

    float* win  = lds_in  + wave * 64;
    float* wout = lds_out + wave * 512;

    // ---- Phase 1: stage 4 elements' params into LDS (coalesced b64 loads) ----
    {
        const int e_ld = base_e + (lane >> 3);   // element owning this lane's float2
        const int off  = lane * 2;
        float2 v = make_float2(0.f, 0.f);
        if (e_ld < B) v = *(const float2*)(x + (size_t)base_e * 16 + off);
        win[off]     = v.x;                      // adjacent + 8B aligned -> ds_store_b64
        win[off + 1] = v.y;
    }
    wave_lds_fence();

    // ---- Phase 2: branchless per-lane fragment gather ------------------------
    // A layout (16x4 f32): lane l<16 holds K=0,1; lane l+16 holds K=2,3.
    // B (4x16) analogous; for A=L, B=L^T the per-lane fragments coincide.
    const int h = lane >> 4;      // selects K pair {2h, 2h+1}
    const int r = lane & 15;      // A row / B column
    const int e = r >> 2;         // element within the wave's group of 4
    const int t = r & 3;          // row of L within that element
    const float* xe = win + e * 16;

    // flat param layout: row t starts at t*t; off-diag (t,c): re=t*t+2c, im=+1; diag at t*t+2t
    // Single softplus per lane (at most one of this lane's two columns is diagonal).
    const float dv = xe[t * t + 2 * t];
    const float sp = fmaxf(dv, 0.f) + log1pf(expf(-fabsf(dv)));

    v2f lr, li;
    #pragma unroll
    for (int q = 0; q < 2; ++q) {
        const int   c      = 2 * h + q;
        const int   cc     = (c < t) ? c : t;          // clamped, always in-bounds
        const int   idx    = t * t + 2 * cc;
        const float re_raw = xe[idx];                  // unconditional adjacent pair
        const float im_raw = xe[idx + 1];              //   -> ds_load_2addr_b32
        const float re     = (c > t) ? 0.f : ((c == t) ? sp : re_raw);
        const float im     = (c < t) ? im_raw : 0.f;
        if (q == 0) { lr.x = re; li.x = im; }
        else        { lr.y = re; li.y = im; }
    }

    // ---- Phase 3: 4x v_wmma_f32_16x16x4_f32 ----------------------------------
    v8f zero8 = {};
    v8f acc_r = __builtin_amdgcn_wmma_f32_16x16x4_f32(
        false, lr, false, lr, (short)0, zero8, false, false);      // Lr Lr^T
    acc_r = __builtin_amdgcn_wmma_f32_16x16x4_f32(
        false, li, false, li, (short)0, acc_r, false, false);      // + Li Li^T
    v8f t1 = __builtin_amdgcn_wmma_f32_16x16x4_f32(
        false, li, false, lr, (short)0, zero8, false, false);      // Li Lr^T
    v8f t2 = __builtin_amdgcn_wmma_f32_16x16x4_f32(
        false, lr, false, li, (short)0, zero8, false, false);      // Lr Li^T

    // ---- Phase 4: store the FULL 16x16 D tile, no selection ------------------
    // Row-major [M][N] of (re,im): lane owns column N=r of rows M=8h+v.
    // 8 ds_store_b64 at constant offsets -- pure straight-line code.
    {
        float* dst = wout + (h * 8) * 32 + r * 2;
        #pragma unroll
        for (int v = 0; v < 8; ++v) {
            *(float2*)(dst + v * 32) = make_float2(acc_r[v], t1[v] - t2[v]);
        }
    }
    wave_lds_fence();

    // ---- Phase 5: gather diagonal block, trace-normalize, b128 stores --------
    {
        const int e2 = lane >> 3;                    // element this lane stores
        const int q  = lane & 7;                     // which float4 of 32 floats
        const int eg = base_e + e2;
        if (eg < B) {
            // element block: row i, col j at e2*136 + i*32 + j*2 (+c)
            const float* blk = wout + e2 * 136;
            const float  tr  = blk[0] + blk[34] + blk[68] + blk[102]; // diag re
            const float  inv = 1.0f / (tr + 1e-8f);
            const int i  = q >> 1;
            const int j0 = (2 * q)     & 3;
            const int j1 = (2 * q + 1) & 3;
            const float2 p0 = *(const float2*)(blk + i * 32 + j0 * 2);
            const float2 p1 = *(const float2*)(blk + i * 32 + j1 * 2);
            const float4 vv = make_float4(p0.x * inv, p0.y * inv,
                                          p1.x * inv, p1.y * inv);
            *(float4*)(out + (size_t)base_e * 32 + lane * 4) = vv;   // coalesced
        }
    }
}

extern "C" void kernel_launch(void* const* d_in, const int* in_sizes, int n_in,
                              void* d_out, int out_size, void* d_ws, size_t ws_size,
                              hipStream_t stream) {
    (void)n_in; (void)out_size; (void)d_ws; (void)ws_size;
    const float* x = (const float*)d_in[0];
    float* out     = (float*)d_out;
    const int B    = in_sizes[0] / 16;
    const int grid = (B + ELEMS_PER_BLOCK - 1) / ELEMS_PER_BLOCK;
    hipLaunchKernelGGL(cholesky_rho_wmma_kernel, dim3(grid), dim3(256), 0, stream,
                       x, out, B);
}